// SoftCLT_Loss_10213432230398
// MI455X (gfx1250) — compile-verified
//
#include <hip/hip_runtime.h>
#include <hip/hip_bf16.h>
#include <math.h>

typedef __attribute__((ext_vector_type(2))) float v2f;
typedef __attribute__((ext_vector_type(8))) float v8f;

#define TWO_BN 4096
#define BN     2048
#define TDIM   96
#define DDIM   64

// ---------- transpose original_series (B,T,N) -> seriesT (B*N, T) ----------
__global__ void k_transpose(const float* __restrict__ series, float* __restrict__ seriesT) {
    int idx = blockIdx.x * 256 + threadIdx.x;                 // over 16*96*128
    if (idx < 16 * 96 * 128) {
        int n = idx & 127;
        int t = (idx >> 7) % 96;
        int b = idx / (96 * 128);
        seriesT[(size_t)(b * 128 + n) * TDIM + t] = series[idx];
    }
}

// ---------- per-row sum of squares of seriesT ----------
__global__ void k_rowss(const float* __restrict__ seriesT, float* __restrict__ ss) {
    int i = blockIdx.x * 256 + threadIdx.x;
    if (i < BN) {
        const float* r = seriesT + (size_t)i * TDIM;
        float s = 0.f;
        #pragma unroll
        for (int t = 0; t < TDIM; ++t) s += r[t] * r[t];
        ss[i] = s;
    }
}

// ---------- sim = F @ F^T via v_wmma_f32_16x16x4_f32 (exact fp32) ----------
// block: (32 lanes, 8 waves); each wave computes one 16x16 tile.
// grid: (32, 256)  -> cols 32*8*16 = 4096, rows 256*16 = 4096
__global__ void k_sim(const float* __restrict__ orig, const float* __restrict__ aug,
                      float* __restrict__ sim) {
    const int lane = threadIdx.x;                    // 0..31
    const int r0 = blockIdx.y * 16;
    const int c0 = (blockIdx.x * 8 + threadIdx.y) * 16;
    const int ar = r0 + (lane & 15);
    const int br = c0 + (lane & 15);
    const float* Arow = (ar < BN) ? (orig + (size_t)ar * DDIM) : (aug + (size_t)(ar - BN) * DDIM);
    const float* Brow = (br < BN) ? (orig + (size_t)br * DDIM) : (aug + (size_t)(br - BN) * DDIM);
    const int kp = (lane >> 4) * 2;                  // K sub-pair per half-wave
    v8f acc = {};
    #pragma unroll
    for (int k = 0; k < DDIM; k += 4) {
        v2f a = *(const v2f*)(Arow + k + kp);        // A[m][k+kp], A[m][k+kp+1]
        v2f b = *(const v2f*)(Brow + k + kp);        // B tile = F^T: same gather on cols
        acc = __builtin_amdgcn_wmma_f32_16x16x4_f32(false, a, false, b,
                                                    (short)0, acc, false, false);
    }
    const int n  = c0 + (lane & 15);
    const int mb = r0 + (lane >> 4) * 8;
    #pragma unroll
    for (int r = 0; r < 8; ++r)
        sim[(size_t)(mb + r) * TWO_BN + n] = acc[r];
}

// ---------- soft = sigmoid(-0.5 * dist), dist from series gram ----------
// grid: (16, 128) -> cols 16*8*16 = 2048, rows 128*16 = 2048
__global__ void k_soft(const float* __restrict__ seriesT, const float* __restrict__ ss,
                       float* __restrict__ soft) {
    const int lane = threadIdx.x;
    const int r0 = blockIdx.y * 16;
    const int c0 = (blockIdx.x * 8 + threadIdx.y) * 16;
    const float* Arow = seriesT + (size_t)(r0 + (lane & 15)) * TDIM;
    const float* Brow = seriesT + (size_t)(c0 + (lane & 15)) * TDIM;
    const int kp = (lane >> 4) * 2;
    v8f acc = {};
    #pragma unroll
    for (int k = 0; k < TDIM; k += 4) {
        v2f a = *(const v2f*)(Arow + k + kp);
        v2f b = *(const v2f*)(Brow + k + kp);
        acc = __builtin_amdgcn_wmma_f32_16x16x4_f32(false, a, false, b,
                                                    (short)0, acc, false, false);
    }
    const int n  = c0 + (lane & 15);
    const int mb = r0 + (lane >> 4) * 8;
    const float ssn = ss[n];
    #pragma unroll
    for (int r = 0; r < 8; ++r) {
        int mrow = mb + r;
        float d = ss[mrow] + ssn - 2.0f * acc[r];
        float e = expf(-0.5f * d);                   // stable: d >= 0 => e <= 1
        soft[(size_t)mrow * BN + n] = e / (1.0f + e);
    }
}

// ---------- per-row: max -> logsumexp -> (Sum s, Sum d*s) -> rowLoss ----------
__global__ void k_row(const float* __restrict__ sim, const float* __restrict__ soft,
                      float* __restrict__ rowLoss) {
    const int i   = blockIdx.x;                      // 0..4095
    const int tid = threadIdx.x;                     // 0..255
    const float* srow    = sim  + (size_t)i * TWO_BN;
    const float* softrow = soft + (size_t)(i & (BN - 1)) * BN;
    __shared__ float red[256];
    __shared__ float shM, shLse, shS1;

    // pass 1: row max of dropped matrix d[i,j]
    float m = -INFINITY;
    for (int j = tid; j < TWO_BN - 1; j += 256) {
        float d = 0.f;
        if (j < i)      d += srow[j];
        if (j >= i - 2) d += srow[j + 1];            // diagonal leaks in at j=i-1
        m = fmaxf(m, d);
    }
    red[tid] = m; __syncthreads();
    for (int off = 128; off > 0; off >>= 1) {
        if (tid < off) red[tid] = fmaxf(red[tid], red[tid + off]);
        __syncthreads();
    }
    if (tid == 0) shM = red[0];
    __syncthreads();
    const float M = shM;

    // pass 2: logsumexp
    float s = 0.f;
    for (int j = tid; j < TWO_BN - 1; j += 256) {
        float d = 0.f;
        if (j < i)      d += srow[j];
        if (j >= i - 2) d += srow[j + 1];
        s += expf(d - M);
    }
    red[tid] = s; __syncthreads();
    for (int off = 128; off > 0; off >>= 1) {
        if (tid < off) red[tid] += red[tid + off];
        __syncthreads();
    }
    if (tid == 0) shLse = logf(red[0]) + M;
    __syncthreads();

    // pass 3: S1 = sum s_ij ; S2 = sum d_ij * s_ij
    float s1 = 0.f, s2 = 0.f;
    for (int j = tid; j < TWO_BN - 1; j += 256) {
        float d = 0.f, sv = 0.f;
        if (j < i)      { d += srow[j];     sv += softrow[j & (BN - 1)]; }
        if (j >= i - 2) { d += srow[j + 1]; sv += softrow[(j + 1) & (BN - 1)]; }
        s1 += sv;
        s2 += d * sv;
    }
    red[tid] = s1; __syncthreads();
    for (int off = 128; off > 0; off >>= 1) {
        if (tid < off) red[tid] += red[tid + off];
        __syncthreads();
    }
    if (tid == 0) shS1 = red[0];
    __syncthreads();
    red[tid] = s2; __syncthreads();
    for (int off = 128; off > 0; off >>= 1) {
        if (tid < off) red[tid] += red[tid + off];
        __syncthreads();
    }
    if (tid == 0) rowLoss[i] = shLse * shS1 - red[0];
}

// ---------- deterministic chunk sums of soft (4096 chunks x 1024 elems) ----------
__global__ void k_chunk_sum(const float* __restrict__ soft, float* __restrict__ partial) {
    const int b   = blockIdx.x;
    const int tid = threadIdx.x;
    __shared__ float red[256];
    const float* p = soft + (size_t)b * 1024;
    float s = 0.f;
    for (int q = tid; q < 1024; q += 256) s += p[q];
    red[tid] = s; __syncthreads();
    for (int off = 128; off > 0; off >>= 1) {
        if (tid < off) red[tid] += red[tid + off];
        __syncthreads();
    }
    if (tid == 0) partial[b] = red[0];
}

// ---------- final: reduce rowLoss and partial soft sums to 2 outputs ----------
__global__ void k_final(const float* __restrict__ rowLoss, const float* __restrict__ partialSoft,
                        float* __restrict__ out) {
    const int tid = threadIdx.x;
    __shared__ float red[256];
    __shared__ float shLoss;

    float s = 0.f;
    for (int q = tid; q < TWO_BN; q += 256) s += rowLoss[q];
    red[tid] = s; __syncthreads();
    for (int off = 128; off > 0; off >>= 1) {
        if (tid < off) red[tid] += red[tid + off];
        __syncthreads();
    }
    if (tid == 0) shLoss = red[0];
    __syncthreads();

    s = 0.f;
    for (int q = tid; q < TWO_BN; q += 256) s += partialSoft[q];
    red[tid] = s; __syncthreads();
    for (int off = 128; off > 0; off >>= 1) {
        if (tid < off) red[tid] += red[tid + off];
        __syncthreads();
    }
    if (tid == 0) {
        out[0] = shLoss * (1.0f / (4096.0f * 4095.0f));   // mean over (2BN, 2BN-1)
        out[1] = red[0] * (1.0f / (2048.0f * 2048.0f));   // 2x2 tiling preserves mean
    }
}

extern "C" void kernel_launch(void* const* d_in, const int* in_sizes, int n_in,
                              void* d_out, int out_size, void* d_ws, size_t ws_size,
                              hipStream_t stream) {
    (void)in_sizes; (void)n_in; (void)out_size; (void)ws_size;
    const float* orig_f = (const float*)d_in[0];  // (16,128,64)
    const float* aug_f  = (const float*)d_in[1];  // (16,128,64)
    const float* orig_s = (const float*)d_in[2];  // (16,96,128)
    // d_in[3] (augmented_series) is unused by the reference.
    float* out = (float*)d_out;

    char* ws = (char*)d_ws;
    float* sim     = (float*)(ws);                                            // 64 MB
    float* soft    = (float*)(ws + 67108864);                                 // 16 MB
    float* seriesT = (float*)(ws + 67108864 + 16777216);                      // 768 KB
    float* ss      = (float*)(ws + 67108864 + 16777216 + 786432);             // 8 KB
    float* rowLoss = (float*)(ws + 67108864 + 16777216 + 786432 + 8192);      // 16 KB
    float* partial = (float*)(ws + 67108864 + 16777216 + 786432 + 8192 + 16384); // 16 KB

    k_transpose<<<768, 256, 0, stream>>>(orig_s, seriesT);
    k_rowss<<<8, 256, 0, stream>>>(seriesT, ss);
    k_sim<<<dim3(32, 256), dim3(32, 8), 0, stream>>>(orig_f, aug_f, sim);
    k_soft<<<dim3(16, 128), dim3(32, 8), 0, stream>>>(seriesT, ss, soft);
    k_row<<<TWO_BN, 256, 0, stream>>>(sim, soft, rowLoss);
    k_chunk_sum<<<TWO_BN, 256, 0, stream>>>(soft, partial);
    k_final<<<1, 256, 0, stream>>>(rowLoss, partial, out);
}